// NN_pricing_GNN_4363686773160
// MI455X (gfx1250) — compile-verified
//
#include <hip/hip_runtime.h>
#include <hip/hip_bf16.h>

#define HID 64
#define NLAYERS 3
#define BN_EPS 1e-5f

typedef __attribute__((ext_vector_type(2))) float v2f;
typedef __attribute__((ext_vector_type(8))) float v8f;

__device__ __forceinline__ float elu1(float x) {
    return x > 0.0f ? x : (__expf(x) - 1.0f);
}

// ---------------------------------------------------------------- utilities
__global__ void k_fill(float* __restrict__ p, float v, int n) {
    int i = blockIdx.x * blockDim.x + threadIdx.x;
    if (i < n) p[i] = v;
}

// deg[col[e]] += 1  (deg pre-initialized to 1.0 for the self-loop)
__global__ void k_deg(const int* __restrict__ col, float* __restrict__ deg, int E) {
    int e = blockIdx.x * blockDim.x + threadIdx.x;
    if (e < E) atomicAdd(&deg[col[e]], 1.0f);
}

// deg -> dinv in place
__global__ void k_dinv(float* __restrict__ deg, int n) {
    int i = blockIdx.x * blockDim.x + threadIdx.x;
    if (i < n) {
        float d = deg[i];
        deg[i] = d > 0.0f ? rsqrtf(d) : 0.0f;
    }
}

// y[n, c] = b[c] + sum_k x[n,k] * w[k,c]   (F_IN = 4)
__global__ void k_inproj(const float* __restrict__ x, const float* __restrict__ w,
                         const float* __restrict__ b, float* __restrict__ y, int n) {
    int gid = blockIdx.x * blockDim.x + threadIdx.x;
    if (gid >= n * HID) return;
    int node = gid >> 6, c = gid & 63;
    float4 xv = *(const float4*)(x + (size_t)node * 4);
    y[gid] = b[c] + xv.x * w[c] + xv.y * w[HID + c] + xv.z * w[2 * HID + c] + xv.w * w[3 * HID + c];
}

// partial sums / sums-of-squares per channel, atomically accumulated
__global__ void k_bn_stats(const float* __restrict__ y, float* __restrict__ sums,
                           float* __restrict__ sumsq, int n) {
    __shared__ float s1[256];
    __shared__ float s2[256];
    int tid = threadIdx.x;
    int c = tid & 63, r = tid >> 6;
    float a = 0.0f, b = 0.0f;
    for (int node = blockIdx.x * 4 + r; node < n; node += gridDim.x * 4) {
        float v = y[(size_t)node * HID + c];
        a += v;
        b += v * v;
    }
    s1[tid] = a; s2[tid] = b;
    __syncthreads();
    if (tid < 64) {
        a = s1[tid] + s1[tid + 64] + s1[tid + 128] + s1[tid + 192];
        b = s2[tid] + s2[tid + 64] + s2[tid + 128] + s2[tid + 192];
        atomicAdd(&sums[tid], a);
        atomicAdd(&sumsq[tid], b);
    }
}

__global__ void k_bn_finalize(const float* __restrict__ sums, const float* __restrict__ sumsq,
                              const float* __restrict__ g, const float* __restrict__ bb,
                              float* __restrict__ scale, float* __restrict__ shift, float inv_n) {
    int c = threadIdx.x;
    if (c < HID) {
        float m = sums[c] * inv_n;
        float v = sumsq[c] * inv_n - m * m;
        float s = g[c] * rsqrtf(v + BN_EPS);
        scale[c] = s;
        shift[c] = bb[c] - m * s;
    }
}

__global__ void k_bn_apply_elu(float* __restrict__ y, const float* __restrict__ scale,
                               const float* __restrict__ shift, int total) {
    int gid = blockIdx.x * blockDim.x + threadIdx.x;
    if (gid < total) {
        int c = gid & 63;
        y[gid] = elu1(scale[c] * y[gid] + shift[c]);
    }
}

// ---------------------------------------------------------------- WMMA GEMM
// out[N,64] = h[N,64] @ W[64,64], fp32 via V_WMMA_F32_16X16X4_F32.
// Block = 256 threads = 8 waves: wave covers tile (m_tile, n_tile) with
// m_tile = blockIdx.x*2 + (wave>>2), n_tile = wave&3 (16 cols each).
__global__ void __launch_bounds__(256)
k_gemm_wmma(const float* __restrict__ h, const float* __restrict__ W,
            float* __restrict__ out, int n) {
    __shared__ float Wl[HID * HID];
    const int tid = threadIdx.x;
    for (int i = tid; i < HID * HID; i += 256) Wl[i] = W[i];
    __syncthreads();

    const int wave = tid >> 5;
    const int lane = tid & 31;
    const int mtiles = (n + 15) >> 4;
    const int m_tile = blockIdx.x * 2 + (wave >> 2);
    if (m_tile >= mtiles) return;              // wave-uniform: EXEC stays all-ones
    const int n0 = (wave & 3) * 16;

    // A fragment addressing (32-bit A 16x4): lanes 0-15 -> M, VGPR0 holds K=0|2, VGPR1 K=1|3
    const int mrow = lane & 15;
    const int kb = (lane >> 4) << 1;           // 0 for lanes 0-15, 2 for lanes 16-31
    int node = m_tile * 16 + mrow;
    if (node > n - 1) node = n - 1;            // clamp keeps EXEC full
    const float* hrow = h + (size_t)node * HID;
    const int ncol = n0 + (lane & 15);         // B fragment: lanes 0-15 -> N

    v8f c = {};
#pragma unroll
    for (int k0 = 0; k0 < HID; k0 += 4) {
        v2f a, b;
        a.x = hrow[k0 + kb];
        a.y = hrow[k0 + kb + 1];
        b.x = Wl[(k0 + kb) * HID + ncol];
        b.y = Wl[(k0 + kb + 1) * HID + ncol];
        c = __builtin_amdgcn_wmma_f32_16x16x4_f32(false, a, false, b, (short)0, c,
                                                  false, false);
    }

    // D layout: VGPR i -> M = i + 8*lane[4], N = lane&15
    const int mbase = m_tile * 16 + ((lane >> 4) << 3);
    const int ccol = lane & 15;
#pragma unroll
    for (int i = 0; i < 8; ++i) {
        int m = mbase + i;
        if (m < n) out[(size_t)m * HID + n0 + ccol] = c[i];
    }
}

// ---------------------------------------------------------------- aggregation
// agg[n,c] = conv_b[c] + dinv[n]^2 * hw[n,c]   (self-loop term + bias)
__global__ void k_agg_init(const float* __restrict__ hw, const float* __restrict__ dinv,
                           const float* __restrict__ bias, float* __restrict__ agg, int n) {
    int gid = blockIdx.x * blockDim.x + threadIdx.x;
    if (gid >= n * HID) return;
    int node = gid >> 6, c = gid & 63;
    float di = dinv[node];
    agg[gid] = bias[c] + di * di * hw[gid];
}

// agg[col[e], c] += dinv[row]*dinv[col] * hw[row[e], c]
__global__ void k_edge_scatter(const int* __restrict__ row, const int* __restrict__ col,
                               const float* __restrict__ dinv, const float* __restrict__ hw,
                               float* __restrict__ agg, int E) {
    int gid = blockIdx.x * blockDim.x + threadIdx.x;
    int e = gid >> 6;
    if (e >= E) return;
    int c = gid & 63;
    int r = row[e], t = col[e];
    float v = dinv[r] * dinv[t] * hw[(size_t)r * HID + c];
    atomicAdd(&agg[(size_t)t * HID + c], v);
}

// ---------------------------------------------------------------- output head
// out[n] = b_out + sum_{l,j} xs[l][n,j] * w_out[l*64+j]
__global__ void k_output(const float* __restrict__ xs, const float* __restrict__ wout,
                         const float* __restrict__ bout, float* __restrict__ out, int n) {
    int i = blockIdx.x * blockDim.x + threadIdx.x;
    if (i >= n) return;
    float acc = bout[0];
#pragma unroll
    for (int l = 0; l < NLAYERS; ++l) {
        const float* p = xs + ((size_t)l * n + i) * HID;
        const float* w = wout + l * HID;
#pragma unroll
        for (int j = 0; j < HID; ++j) acc += p[j] * w[j];
    }
    out[i] = acc;
}

// ---------------------------------------------------------------- launcher
extern "C" void kernel_launch(void* const* d_in, const int* in_sizes, int n_in,
                              void* d_out, int out_size, void* d_ws, size_t ws_size,
                              hipStream_t stream) {
    const float* x      = (const float*)d_in[0];
    const int*   ei     = (const int*)  d_in[1];
    const float* w_in   = (const float*)d_in[2];
    const float* b_in   = (const float*)d_in[3];
    const float* bn0_g  = (const float*)d_in[4];
    const float* bn0_b  = (const float*)d_in[5];
    const float* conv_W = (const float*)d_in[6];
    const float* conv_b = (const float*)d_in[7];
    const float* bn_g   = (const float*)d_in[8];
    const float* bn_b   = (const float*)d_in[9];
    const float* w_out  = (const float*)d_in[10];
    const float* b_out  = (const float*)d_in[11];
    float* out = (float*)d_out;

    const int N = in_sizes[0] / 4;       // F_IN = 4
    const int E = in_sizes[1] / 2;
    const int* row = ei;                 // edge_index[0] = source
    const int* col = ei + E;             // edge_index[1] = target

    // workspace carve-up (floats)
    float* ws = (float*)d_ws;
    size_t off = 0;
    float* dinv   = ws + off; off += ((size_t)N + 63) & ~(size_t)63;
    float* buf_h  = ws + off; off += (size_t)N * HID;
    float* buf_hw = ws + off; off += (size_t)N * HID;
    float* xs     = ws + off; off += (size_t)NLAYERS * N * HID;
    float* sums   = ws + off; off += 64;
    float* sumsq  = ws + off; off += 64;   // contiguous with sums
    float* scale  = ws + off; off += 64;
    float* shift  = ws + off; off += 64;
    (void)sumsq; (void)ws_size; (void)n_in; (void)out_size;

    const int T = 256;
    auto cdiv = [](int a, int b) { return (a + b - 1) / b; };
    const float inv_n = 1.0f / (float)N;

    // GCN normalization: deg (with self-loop) -> dinv
    k_fill<<<cdiv(N, T), T, 0, stream>>>(dinv, 1.0f, N);
    k_deg<<<cdiv(E, T), T, 0, stream>>>(col, dinv, E);
    k_dinv<<<cdiv(N, T), T, 0, stream>>>(dinv, N);

    // input projection -> BN0 -> ELU
    k_inproj<<<cdiv(N * HID, T), T, 0, stream>>>(x, w_in, b_in, buf_h, N);
    k_fill<<<1, 128, 0, stream>>>(sums, 0.0f, 128);   // zeros sums+sumsq
    k_bn_stats<<<128, T, 0, stream>>>(buf_h, sums, sumsq, N);
    k_bn_finalize<<<1, 64, 0, stream>>>(sums, sumsq, bn0_g, bn0_b, scale, shift, inv_n);
    k_bn_apply_elu<<<cdiv(N * HID, T), T, 0, stream>>>(buf_h, scale, shift, N * HID);

    const int mtiles = cdiv(N, 16);
    const float* hin = buf_h;
    for (int l = 0; l < NLAYERS; ++l) {
        float* hl = xs + (size_t)l * N * HID;     // layer output goes straight into JK slot
        k_gemm_wmma<<<cdiv(mtiles, 2), T, 0, stream>>>(hin, conv_W + (size_t)l * HID * HID,
                                                       buf_hw, N);
        k_agg_init<<<cdiv(N * HID, T), T, 0, stream>>>(buf_hw, dinv, conv_b + l * HID, hl, N);
        k_edge_scatter<<<cdiv(E * HID, T), T, 0, stream>>>(row, col, dinv, buf_hw, hl, E);
        k_fill<<<1, 128, 0, stream>>>(sums, 0.0f, 128);
        k_bn_stats<<<128, T, 0, stream>>>(hl, sums, sumsq, N);
        k_bn_finalize<<<1, 64, 0, stream>>>(sums, sumsq, bn_g + l * HID, bn_b + l * HID,
                                            scale, shift, inv_n);
        k_bn_apply_elu<<<cdiv(N * HID, T), T, 0, stream>>>(hl, scale, shift, N * HID);
        hin = hl;
    }

    k_output<<<cdiv(N, T), T, 0, stream>>>(xs, w_out, b_out, out, N);
}